// TrojanDetector_22230750724572
// MI455X (gfx1250) — compile-verified
//
#include <hip/hip_runtime.h>

#define NNODES 50000
#define NEDGES 800000
#define HDIM   128
#define INDIM  256
#define NLAYERS 4
#define NGRAPH 64
#define BN_EPS 1e-5f

typedef float v2f __attribute__((ext_vector_type(2)));
typedef float v8f __attribute__((ext_vector_type(8)));

// ---------------------------------------------------------------------------
// GEMM: Y[M,128] = (A [+ Adds]) [M,K] @ W[K,128] + bias, optional ReLU.
// One wave -> 32 rows x 128 cols via V_WMMA_F32_16X16X4_F32 (2x8 C tiles),
// reusing each B fragment for two WMMAs (halves LDS traffic per FLOP).
// Block = 256 threads (8 waves) -> 256 rows. W staged in LDS pair-interleaved:
//   Wl[kpair*256 + ((n*2 + (k&1)) ^ ((kpair&1)*32))]
// so a B fragment is ONE contiguous ds_load_b64, and the XOR swizzle puts the
// two half-waves on disjoint 32-bank sets (kpair parity == half-wave id g2).
// ---------------------------------------------------------------------------
template <bool HAS_ADD>
__global__ void __launch_bounds__(256)
gin_gemm_wmma(const float* __restrict__ A, const float* __restrict__ Adds,
              const float* __restrict__ W, const float* __restrict__ bias,
              float* __restrict__ Y, int M, int K, int relu_out)
{
    __shared__ float Wl[HDIM * HDIM];            // 64 KB (of 320 KB / WGP)

    const int tid  = threadIdx.x;
    const int wave = tid >> 5;
    const int lane = tid & 31;
    const int l16  = lane & 15;
    const int g2   = lane >> 4;                  // half-wave select
    const int rowBase = blockIdx.x * 256 + wave * 32;

    // A-fragment rows for this lane (clamped; tail rows are masked at store)
    int arow0 = rowBase + l16;       arow0 = arow0 < M ? arow0 : M - 1;
    int arow1 = rowBase + 16 + l16;  arow1 = arow1 < M ? arow1 : M - 1;
    const float* __restrict__ Arow0 = A + (size_t)arow0 * K + 2 * g2;
    const float* __restrict__ Arow1 = A + (size_t)arow1 * K + 2 * g2;
    const float* __restrict__ Drow0 = HAS_ADD ? (Adds + (size_t)arow0 * K + 2 * g2)
                                              : nullptr;
    const float* __restrict__ Drow1 = HAS_ADD ? (Adds + (size_t)arow1 * K + 2 * g2)
                                              : nullptr;
    __builtin_prefetch(Arow0, 0, 0);             // global_prefetch_b8
    __builtin_prefetch(Arow1, 0, 0);
    if (HAS_ADD) { __builtin_prefetch(Drow0, 0, 0); __builtin_prefetch(Drow1, 0, 0); }

    // LDS base for B fragments: + g2*256 selects the odd kpair panel
    const float* __restrict__ Bbase = Wl + g2 * 256 + l16 * 2;

    v8f acc0[8] = {};
    v8f acc1[8] = {};

    for (int kc = 0; kc < K; kc += HDIM) {
        __syncthreads();
        // stage W[kc..kc+127][:] pair-interleaved + XOR-swizzled into LDS
        #pragma unroll
        for (int i = 0; i < 16; ++i) {
            const int u  = i * 256 + tid;
            const int kp = u >> 6;               // k pair 0..63
            const int np = u & 63;               // column pair 0..63
            const float* wp = W + (size_t)(kc + kp * 2) * HDIM + np * 2;
            v2f lo = *(const v2f*)wp;            // row k,   cols n,n+1
            v2f hi = *(const v2f*)(wp + HDIM);   // row k+1, cols n,n+1
            float4 q; q.x = lo.x; q.y = hi.x; q.z = lo.y; q.w = hi.y;
            *(float4*)(Wl + kp * 256 + ((np ^ ((kp & 1) << 3)) << 2)) = q;
        }
        __syncthreads();

        #pragma unroll 2
        for (int ks = 0; ks < HDIM; ks += 4) {
            // A fragments: M = l16 (+16), K = kc + ks + 2*g2 + {0,1}
            v2f a0 = *(const v2f*)(Arow0 + kc + ks);
            v2f a1 = *(const v2f*)(Arow1 + kc + ks);
            if (HAS_ADD) {
                v2f d0 = *(const v2f*)(Drow0 + kc + ks);
                v2f d1 = *(const v2f*)(Drow1 + kc + ks);
                a0 = a0 + d0;
                a1 = a1 + d1;
            }
            const float* __restrict__ Bk = Bbase + (ks >> 1) * 256;
            #pragma unroll
            for (int nt = 0; nt < 8; ++nt) {
                // swizzled tile address; data is for column tile nt
                v2f b = *(const v2f*)(Bk + ((nt ^ g2) << 5));
                acc0[nt] = __builtin_amdgcn_wmma_f32_16x16x4_f32(
                    false, a0, false, b, (short)0, acc0[nt], false, false);
                acc1[nt] = __builtin_amdgcn_wmma_f32_16x16x4_f32(
                    false, a1, false, b, (short)0, acc1[nt], false, false);
            }
        }
    }

    // epilogue: VGPR r of tile (h,nt) holds
    //   (M = rowBase + h*16 + r + 8*g2, N = nt*16 + l16)
    #pragma unroll
    for (int nt = 0; nt < 8; ++nt) {
        const int col = nt * 16 + l16;
        const float bv = bias[col];
        #pragma unroll
        for (int r = 0; r < 8; ++r) {
            const int row0 = rowBase + r + 8 * g2;
            if (row0 < M) {
                float v = acc0[nt][r] + bv;
                if (relu_out) v = v > 0.0f ? v : 0.0f;
                Y[(size_t)row0 * HDIM + col] = v;
            }
            const int row1 = rowBase + 16 + r + 8 * g2;
            if (row1 < M) {
                float v = acc1[nt][r] + bv;
                if (relu_out) v = v > 0.0f ? v : 0.0f;
                Y[(size_t)row1 * HDIM + col] = v;
            }
        }
    }
}

// ---------------------------------------------------------------------------
// Edge scatter-add: agg[dst] += x[src].  One wave per edge, float4 per lane,
// native global_atomic_add_f32 (one row = 512B fully coalesced per wave).
// ---------------------------------------------------------------------------
__global__ void __launch_bounds__(256)
gin_scatter(const int* __restrict__ src, const int* __restrict__ dst,
            const float* __restrict__ x, float* __restrict__ agg, int ne)
{
    const int e    = (int)((blockIdx.x * 256 + threadIdx.x) >> 5);
    const int lane = threadIdx.x & 31;
    if (e >= ne) return;
    const int s = src[e];
    const int d = dst[e];
    float4 v = *(const float4*)(x + (size_t)s * HDIM + lane * 4);
    float* p = agg + (size_t)d * HDIM + lane * 4;
    unsafeAtomicAdd(p + 0, v.x);
    unsafeAtomicAdd(p + 1, v.y);
    unsafeAtomicAdd(p + 2, v.z);
    unsafeAtomicAdd(p + 3, v.w);
}

// ---------------------------------------------------------------------------
// Per-column sum / sum-of-squares partial reduction (training-mode BN).
// ---------------------------------------------------------------------------
__global__ void __launch_bounds__(128)
gin_colstats(const float* __restrict__ Y, float* __restrict__ stats,
             int M, int rowsPerBlock)
{
    const int col = threadIdx.x;
    int r0 = blockIdx.x * rowsPerBlock;
    int r1 = r0 + rowsPerBlock; if (r1 > M) r1 = M;
    float s = 0.0f, s2 = 0.0f;
    for (int r = r0; r < r1; ++r) {
        float v = Y[(size_t)r * HDIM + col];
        s += v; s2 += v * v;
    }
    unsafeAtomicAdd(&stats[col], s);
    unsafeAtomicAdd(&stats[HDIM + col], s2);
}

__global__ void gin_scaleshift(const float* __restrict__ stats,
                               const float* __restrict__ gamma,
                               const float* __restrict__ beta,
                               float* __restrict__ scsh, int M)
{
    const int c = threadIdx.x;
    const float inv = 1.0f / (float)M;
    float mu  = stats[c] * inv;
    float var = stats[HDIM + c] * inv - mu * mu;
    float sc  = gamma[c] * __frsqrt_rn(var + BN_EPS);
    scsh[c]        = sc;
    scsh[HDIM + c] = beta[c] - mu * sc;
}

// out = relu(Y * scale + shift), float4-vectorized.
__global__ void __launch_bounds__(256)
gin_bnapply(const float* __restrict__ Y, const float* __restrict__ scsh,
            float* __restrict__ X, int M)
{
    const long idx = (long)blockIdx.x * 256 + threadIdx.x;     // float4 index
    const long total = (long)M * (HDIM / 4);
    if (idx >= total) return;
    const int cb = (int)(idx & (HDIM / 4 - 1));
    float4 y  = ((const float4*)Y)[idx];
    float4 sc = ((const float4*)scsh)[cb];
    float4 sh = ((const float4*)(scsh + HDIM))[cb];
    float4 o;
    o.x = fmaxf(y.x * sc.x + sh.x, 0.0f);
    o.y = fmaxf(y.y * sc.y + sh.y, 0.0f);
    o.z = fmaxf(y.z * sc.z + sh.z, 0.0f);
    o.w = fmaxf(y.w * sc.w + sh.w, 0.0f);
    ((float4*)X)[idx] = o;
}

// ---------------------------------------------------------------------------
// Global mean pool (segment sums + counts) and the 2-layer MLP head.
// ---------------------------------------------------------------------------
__global__ void __launch_bounds__(128)
gin_pool(const float* __restrict__ x, const int* __restrict__ batch,
         float* __restrict__ pooled, float* __restrict__ cnt,
         int M, int rowsPerBlock)
{
    const int t = threadIdx.x;
    int r0 = blockIdx.x * rowsPerBlock;
    int r1 = r0 + rowsPerBlock; if (r1 > M) r1 = M;
    for (int r = r0; r < r1; ++r) {
        const int g = batch[r];
        unsafeAtomicAdd(&pooled[g * HDIM + t], x[(size_t)r * HDIM + t]);
        if (t == 0) unsafeAtomicAdd(&cnt[g], 1.0f);
    }
}

__global__ void __launch_bounds__(128)
gin_head(const float* __restrict__ pooled, const float* __restrict__ cnt,
         const float* __restrict__ Wh1, const float* __restrict__ bh1,
         const float* __restrict__ Wh2, const float* __restrict__ bh2,
         float* __restrict__ out)
{
    __shared__ float p[HDIM];
    __shared__ float h[HDIM];
    const int g = blockIdx.x, t = threadIdx.x;
    float c = cnt[g]; c = c > 1.0f ? c : 1.0f;
    p[t] = pooled[g * HDIM + t] / c;
    __syncthreads();
    float a = bh1[t];
    for (int k = 0; k < HDIM; ++k) a += p[k] * Wh1[k * HDIM + t];
    h[t] = fmaxf(a, 0.0f);
    __syncthreads();
    if (t < 2) {
        float o = bh2[t];
        for (int k = 0; k < HDIM; ++k) o += h[k] * Wh2[k * 2 + t];
        out[g * 2 + t] = o;
    }
}

// ---------------------------------------------------------------------------
extern "C" void kernel_launch(void* const* d_in, const int* in_sizes, int n_in,
                              void* d_out, int out_size, void* d_ws, size_t ws_size,
                              hipStream_t stream)
{
    const float* X    = (const float*)d_in[0];
    const int*   ei   = (const int*)  d_in[1];   // (2,E): [0..E)=src, [E..2E)=dst
    const int*   batch= (const int*)  d_in[2];
    const float* Wp   = (const float*)d_in[3];
    const float* bp   = (const float*)d_in[4];
    const float* W1   = (const float*)d_in[5];
    const float* b1   = (const float*)d_in[6];
    const float* g1   = (const float*)d_in[7];
    const float* be1  = (const float*)d_in[8];
    const float* W2   = (const float*)d_in[9];
    const float* b2   = (const float*)d_in[10];
    const float* gbn  = (const float*)d_in[11];
    const float* bbn  = (const float*)d_in[12];
    const float* Wh1  = (const float*)d_in[13];
    const float* bh1  = (const float*)d_in[14];
    const float* Wh2  = (const float*)d_in[15];
    const float* bh2  = (const float*)d_in[16];

    const size_t NH = (size_t)NNODES * HDIM;
    float* ws     = (float*)d_ws;
    float* x      = ws;                 // [N,128] current node features
    float* agg    = x    + NH;          // [N,128] scatter accumulator
    float* bufA   = agg  + NH;          // [N,128] BN(relu) of GEMM1 output
    float* bufB   = bufA + NH;          // [N,128] raw GEMM output
    float* stats  = bufB + NH;          // 256 floats: colsum, colsumsq
    float* scsh   = stats + 2 * HDIM;   // 256 floats: scale, shift
    float* pooled = scsh  + 2 * HDIM;   // [64,128]
    float* cnt    = pooled + NGRAPH * HDIM; // [64]

    const int gemmBlocks = (NNODES + 255) / 256;          // 196 (256 rows/block)
    const int rpb        = (NNODES + 255) / 256;          // rows per stats block
    const int bnBlocks   = (NNODES * (HDIM / 4) + 255) / 256;
    const int scatBlocks = NEDGES / 8;                    // 8 edges (waves)/block

    // input projection: x = relu(X @ Wp + bp), K = 256
    gin_gemm_wmma<false><<<gemmBlocks, 256, 0, stream>>>(X, nullptr, Wp, bp, x,
                                                         NNODES, INDIM, 1);

    for (int l = 0; l < NLAYERS; ++l) {
        hipMemsetAsync(agg, 0, NH * sizeof(float), stream);
        gin_scatter<<<scatBlocks, 256, 0, stream>>>(ei, ei + NEDGES, x, agg, NEDGES);

        // bufB = (x + agg) @ W1[l] + b1[l]   (addend fused into A load)
        gin_gemm_wmma<true><<<gemmBlocks, 256, 0, stream>>>(x, agg,
                                                            W1 + l * HDIM * HDIM,
                                                            b1 + l * HDIM, bufB,
                                                            NNODES, HDIM, 0);
        hipMemsetAsync(stats, 0, 2 * HDIM * sizeof(float), stream);
        gin_colstats<<<256, 128, 0, stream>>>(bufB, stats, NNODES, rpb);
        gin_scaleshift<<<1, 128, 0, stream>>>(stats, g1 + l * HDIM, be1 + l * HDIM,
                                              scsh, NNODES);
        gin_bnapply<<<bnBlocks, 256, 0, stream>>>(bufB, scsh, bufA, NNODES);

        // bufB = bufA @ W2[l] + b2[l]
        gin_gemm_wmma<false><<<gemmBlocks, 256, 0, stream>>>(bufA, nullptr,
                                                             W2 + l * HDIM * HDIM,
                                                             b2 + l * HDIM, bufB,
                                                             NNODES, HDIM, 0);
        hipMemsetAsync(stats, 0, 2 * HDIM * sizeof(float), stream);
        gin_colstats<<<256, 128, 0, stream>>>(bufB, stats, NNODES, rpb);
        gin_scaleshift<<<1, 128, 0, stream>>>(stats, gbn + l * HDIM, bbn + l * HDIM,
                                              scsh, NNODES);
        gin_bnapply<<<bnBlocks, 256, 0, stream>>>(bufB, scsh, x, NNODES);
    }

    hipMemsetAsync(pooled, 0, (NGRAPH * HDIM + NGRAPH) * sizeof(float), stream);
    gin_pool<<<256, 128, 0, stream>>>(x, batch, pooled, cnt, NNODES, rpb);
    gin_head<<<NGRAPH, 128, 0, stream>>>(pooled, cnt, Wh1, bh1, Wh2, bh2,
                                         (float*)d_out);
}